// MultiheadSelfAttention_19078244728944
// MI455X (gfx1250) — compile-verified
//
#include <hip/hip_runtime.h>
#include <math.h>

typedef __attribute__((ext_vector_type(2))) float v2f;
typedef __attribute__((ext_vector_type(8))) float v8f;

#define DIM_B 2
#define DIM_T 2048
#define DIM_K 1024
#define NHEAD 16
#define HDIM  64

static __device__ __forceinline__ v8f wmma_f32(v2f a, v2f b, v8f c) {
  // V_WMMA_F32_16X16X4_F32: D = A(16x4) * B(4x16) + C(16x16), all f32
  return __builtin_amdgcn_wmma_f32_16x16x4_f32(false, a, false, b, (short)0, c,
                                               false, false);
}

// ---------------------------------------------------------------------------
// C[M,N] = A[M,K] * W[N,K]^T (+ bias[n])     (torch Linear: y = x @ W.T + b)
// block: 256 threads = 8 waves; wave tile 64(M) x 32(N): 6 loads : 8 WMMAs.
// Distance-2 software pipeline with two named fragment buffers (no rotation
// copies): each load batch has a full 8-WMMA phase in flight before its
// consumers wait on it. block tile 128 x 128.
// ---------------------------------------------------------------------------
__global__ __launch_bounds__(256)
__attribute__((amdgpu_waves_per_eu(2)))
void gemm_xwt_kernel(const float* __restrict__ A, const float* __restrict__ W,
                     const float* __restrict__ bias, float* __restrict__ C,
                     int M, int N, int K) {
  const int lane  = threadIdx.x & 31;
  const int wave  = threadIdx.x >> 5;
  const int wm    = wave >> 2;              // 0..1
  const int wn    = wave & 3;               // 0..3
  const int m0    = blockIdx.y * 128 + wm * 64;
  const int n0    = blockIdx.x * 128 + wn * 32;
  const int lhalf = lane >> 4;              // which 16-lane half
  const int lmod  = lane & 15;
  const int kbase = lhalf * 2;              // lanes 16-31 hold K=k0+2,k0+3

  v8f acc[4][2];
#pragma unroll
  for (int i = 0; i < 4; ++i)
#pragma unroll
    for (int j = 0; j < 2; ++j)
#pragma unroll
      for (int r = 0; r < 8; ++r) acc[i][j][r] = 0.0f;

  const float* ar0 = A + (size_t)(m0 + lmod) * K;
  const float* ar1 = ar0 + (size_t)16 * K;
  const float* ar2 = ar0 + (size_t)32 * K;
  const float* ar3 = ar0 + (size_t)48 * K;
  const float* br0 = W + (size_t)(n0 + lmod) * K;
  const float* br1 = br0 + (size_t)16 * K;

  // prologue: buffer0 <- step k=0, buffer1 <- step k=4
  v2f a0 = *(const v2f*)(ar0 + kbase);
  v2f a1 = *(const v2f*)(ar1 + kbase);
  v2f a2 = *(const v2f*)(ar2 + kbase);
  v2f a3 = *(const v2f*)(ar3 + kbase);
  v2f b0 = *(const v2f*)(br0 + kbase);
  v2f b1 = *(const v2f*)(br1 + kbase);

  v2f c0 = *(const v2f*)(ar0 + 4 + kbase);
  v2f c1 = *(const v2f*)(ar1 + 4 + kbase);
  v2f c2 = *(const v2f*)(ar2 + 4 + kbase);
  v2f c3 = *(const v2f*)(ar3 + 4 + kbase);
  v2f d0 = *(const v2f*)(br0 + 4 + kbase);
  v2f d1 = *(const v2f*)(br1 + 4 + kbase);

#pragma clang loop unroll(disable)
  for (int k0 = 0; k0 + 16 <= K; k0 += 8) {
    // phase 0: consume buffer0 (step k0), then refill it with step k0+8;
    // the refill flies while phase 1's WMMAs run.
    acc[0][0] = wmma_f32(a0, b0, acc[0][0]);
    acc[0][1] = wmma_f32(a0, b1, acc[0][1]);
    acc[1][0] = wmma_f32(a1, b0, acc[1][0]);
    acc[1][1] = wmma_f32(a1, b1, acc[1][1]);
    acc[2][0] = wmma_f32(a2, b0, acc[2][0]);
    acc[2][1] = wmma_f32(a2, b1, acc[2][1]);
    acc[3][0] = wmma_f32(a3, b0, acc[3][0]);
    acc[3][1] = wmma_f32(a3, b1, acc[3][1]);
    {
      const int kn = k0 + 8 + kbase;
      a0 = *(const v2f*)(ar0 + kn);
      a1 = *(const v2f*)(ar1 + kn);
      a2 = *(const v2f*)(ar2 + kn);
      a3 = *(const v2f*)(ar3 + kn);
      b0 = *(const v2f*)(br0 + kn);
      b1 = *(const v2f*)(br1 + kn);
    }
    // phase 1: consume buffer1 (step k0+4), refill with step k0+12; the
    // refill flies while the next iteration's phase 0 WMMAs run.
    acc[0][0] = wmma_f32(c0, d0, acc[0][0]);
    acc[0][1] = wmma_f32(c0, d1, acc[0][1]);
    acc[1][0] = wmma_f32(c1, d0, acc[1][0]);
    acc[1][1] = wmma_f32(c1, d1, acc[1][1]);
    acc[2][0] = wmma_f32(c2, d0, acc[2][0]);
    acc[2][1] = wmma_f32(c2, d1, acc[2][1]);
    acc[3][0] = wmma_f32(c3, d0, acc[3][0]);
    acc[3][1] = wmma_f32(c3, d1, acc[3][1]);
    {
      const int kn = k0 + 12 + kbase;
      c0 = *(const v2f*)(ar0 + kn);
      c1 = *(const v2f*)(ar1 + kn);
      c2 = *(const v2f*)(ar2 + kn);
      c3 = *(const v2f*)(ar3 + kn);
      d0 = *(const v2f*)(br0 + kn);
      d1 = *(const v2f*)(br1 + kn);
    }
  }
  // epilogue: steps K-8 (buffer0) and K-4 (buffer1)
  acc[0][0] = wmma_f32(a0, b0, acc[0][0]);
  acc[0][1] = wmma_f32(a0, b1, acc[0][1]);
  acc[1][0] = wmma_f32(a1, b0, acc[1][0]);
  acc[1][1] = wmma_f32(a1, b1, acc[1][1]);
  acc[2][0] = wmma_f32(a2, b0, acc[2][0]);
  acc[2][1] = wmma_f32(a2, b1, acc[2][1]);
  acc[3][0] = wmma_f32(a3, b0, acc[3][0]);
  acc[3][1] = wmma_f32(a3, b1, acc[3][1]);
  acc[0][0] = wmma_f32(c0, d0, acc[0][0]);
  acc[0][1] = wmma_f32(c0, d1, acc[0][1]);
  acc[1][0] = wmma_f32(c1, d0, acc[1][0]);
  acc[1][1] = wmma_f32(c1, d1, acc[1][1]);
  acc[2][0] = wmma_f32(c2, d0, acc[2][0]);
  acc[2][1] = wmma_f32(c2, d1, acc[2][1]);
  acc[3][0] = wmma_f32(c3, d0, acc[3][0]);
  acc[3][1] = wmma_f32(c3, d1, acc[3][1]);

#pragma unroll
  for (int mi = 0; mi < 4; ++mi)
#pragma unroll
    for (int ni = 0; ni < 2; ++ni)
#pragma unroll
      for (int r = 0; r < 8; ++r) {
        const int row = m0 + mi * 16 + r + lhalf * 8;
        const int col = n0 + ni * 16 + lmod;
        float v = acc[mi][ni][r];
        if (bias) v += bias[col];
        C[(size_t)row * N + col] = v;
      }
}

// half-wave (16-lane) butterfly reductions: rows of a C-tile live within a half
static __device__ __forceinline__ float hred_max(float v) {
  v = fmaxf(v, __shfl_xor(v, 1));
  v = fmaxf(v, __shfl_xor(v, 2));
  v = fmaxf(v, __shfl_xor(v, 4));
  v = fmaxf(v, __shfl_xor(v, 8));
  return v;
}
static __device__ __forceinline__ float hred_sum(float v) {
  v += __shfl_xor(v, 1);
  v += __shfl_xor(v, 2);
  v += __shfl_xor(v, 4);
  v += __shfl_xor(v, 8);
  return v;
}

// ---------------------------------------------------------------------------
// Flash attention: grid (T/64, NHEAD, B), block 128 = 4 waves (1 per SIMD32;
// waves_per_eu(1) gives the full VGPR file), each wave owns 16 query rows
// for one (b,h). q/k/v are (B*T, K) with the head slice at column h*HDIM.
// The LDS staging tile is wave-private and LDS ops from one wave execute in
// order, so no block barriers are needed.
// ---------------------------------------------------------------------------
__global__ __launch_bounds__(128)
__attribute__((amdgpu_waves_per_eu(1)))
void attn_kernel(const float* __restrict__ Q, const float* __restrict__ Km,
                 const float* __restrict__ V, float* __restrict__ O) {
  __shared__ float pbuf[4][16 * 16];        // per-wave P staging (C->A relayout)

  const int lane  = threadIdx.x & 31;
  const int wave  = threadIdx.x >> 5;       // 0..3
  const int lhalf = lane >> 4;
  const int lmod  = lane & 15;
  const int h  = blockIdx.y;
  const int b  = blockIdx.z;
  const int q0 = blockIdx.x * 64 + wave * 16;

  const size_t base = (size_t)b * DIM_T * DIM_K + (size_t)h * HDIM;

  // Preload Q fragments for all 16 K-steps over S=64 (A-layout)
  v2f qa[16];
  {
    const float* qrow = Q + base + (size_t)(q0 + lmod) * DIM_K;
#pragma unroll
    for (int t = 0; t < 16; ++t)
      qa[t] = *(const v2f*)(qrow + t * 4 + lhalf * 2);
  }

  v8f acc[4];
#pragma unroll
  for (int j = 0; j < 4; ++j)
#pragma unroll
    for (int r = 0; r < 8; ++r) acc[j][r] = 0.0f;

  float mrow[8], lrow[8];
#pragma unroll
  for (int r = 0; r < 8; ++r) { mrow[r] = -INFINITY; lrow[r] = 0.0f; }

  const float scale = 0.03125f;             // 1/sqrt(K) = 1/sqrt(1024)

  for (int key0 = 0; key0 < DIM_T; key0 += 16) {
    // ---- preload all K^T fragments for this key chunk, then the serial
    //      score-accumulation WMMA chain runs with no load stalls ----
    const float* krow = Km + base + (size_t)(key0 + lmod) * DIM_K;
    v2f kb[16];
#pragma unroll
    for (int t = 0; t < 16; ++t)
      kb[t] = *(const v2f*)(krow + t * 4 + lhalf * 2);

    v8f s;
#pragma unroll
    for (int r = 0; r < 8; ++r) s[r] = 0.0f;
#pragma unroll
    for (int t = 0; t < 16; ++t) s = wmma_f32(qa[t], kb[t], s);

    // ---- preload V fragments for this chunk (overlap with softmax math) ----
    v2f vf[4][4];
#pragma unroll
    for (int t = 0; t < 4; ++t) {
      const int kk = t * 4 + lhalf * 2;     // key sub-index within chunk
      const float* vbase = V + base + (size_t)(key0 + kk) * DIM_K + lmod;
#pragma unroll
      for (int j = 0; j < 4; ++j) {
        vf[t][j].x = vbase[j * 16];         // V[key0+kk  ][j*16+lmod]
        vf[t][j].y = vbase[j * 16 + DIM_K]; // V[key0+kk+1][j*16+lmod]
      }
    }

    // ---- online softmax (row stats broadcast within each 16-lane half) ----
    float p[8];
#pragma unroll
    for (int r = 0; r < 8; ++r) {
      const float sv   = s[r] * scale;
      const float rm   = hred_max(sv);
      const float mnew = fmaxf(mrow[r], rm);
      const float corr = __expf(mrow[r] - mnew);
      const float pv   = __expf(sv - mnew);
      lrow[r] = lrow[r] * corr + hred_sum(pv);
      mrow[r] = mnew;
      p[r]    = pv;
#pragma unroll
      for (int j = 0; j < 4; ++j) acc[j][r] *= corr;
    }

    // ---- relayout P from C-layout to A-layout through wave-private LDS ----
#pragma unroll
    for (int r = 0; r < 8; ++r)
      pbuf[wave][(r + lhalf * 8) * 16 + lmod] = p[r];

    // ---- O += P(16x16) * V(16x64): 4 K-steps x 4 column tiles ----
#pragma unroll
    for (int t = 0; t < 4; ++t) {
      const int kk = t * 4 + lhalf * 2;
      v2f pa;
      pa.x = pbuf[wave][lmod * 16 + kk];
      pa.y = pbuf[wave][lmod * 16 + kk + 1];
#pragma unroll
      for (int j = 0; j < 4; ++j)
        acc[j] = wmma_f32(pa, vf[t][j], acc[j]);
    }
  }

  // ---- finalize: divide by row sum, store ----
#pragma unroll
  for (int j = 0; j < 4; ++j)
#pragma unroll
    for (int r = 0; r < 8; ++r) {
      const float ov = acc[j][r] / lrow[r];
      O[base + (size_t)(q0 + r + lhalf * 8) * DIM_K + j * 16 + lmod] = ov;
    }
}

// ---------------------------------------------------------------------------
extern "C" void kernel_launch(void* const* d_in, const int* in_sizes, int n_in,
                              void* d_out, int out_size, void* d_ws,
                              size_t ws_size, hipStream_t stream) {
  (void)in_sizes; (void)n_in; (void)out_size; (void)ws_size;
  const float* x  = (const float*)d_in[0];
  const float* wq = (const float*)d_in[1];
  const float* wk = (const float*)d_in[2];
  const float* wv = (const float*)d_in[3];
  const float* wu = (const float*)d_in[4];
  const float* bu = (const float*)d_in[5];
  float* out = (float*)d_out;

  const size_t MAT = (size_t)DIM_B * DIM_T * DIM_K;   // 4M floats per matrix
  float* qb = (float*)d_ws;
  float* kb = qb + MAT;
  float* vb = kb + MAT;
  float* ab = vb + MAT;                               // 64 MB total scratch

  const int M = DIM_B * DIM_T, N = DIM_K, K = DIM_K;
  dim3 gblk(256);
  dim3 ggrid(N / 128, M / 128);

  gemm_xwt_kernel<<<ggrid, gblk, 0, stream>>>(x, wq, nullptr, qb, M, N, K);
  gemm_xwt_kernel<<<ggrid, gblk, 0, stream>>>(x, wk, nullptr, kb, M, N, K);
  gemm_xwt_kernel<<<ggrid, gblk, 0, stream>>>(x, wv, nullptr, vb, M, N, K);

  dim3 agrid(DIM_T / 64, NHEAD, DIM_B);
  attn_kernel<<<agrid, dim3(128), 0, stream>>>(qb, kb, vb, ab);

  gemm_xwt_kernel<<<ggrid, gblk, 0, stream>>>(ab, wu, bu, out, M, N, K);
}